// GNNActor_7301444403652
// MI455X (gfx1250) — compile-verified
//
#include <hip/hip_runtime.h>
#include <hip/hip_bf16.h>

#define N_NODES   50000
#define N_EDGES   400000
#define NODE_FEAT 32
#define HIDDEN    256
#define ACTION_DIM 6
#define NUM_GRAPHS 64

typedef float v2f __attribute__((ext_vector_type(2)));
typedef float v8f __attribute__((ext_vector_type(8)));

// ---------------- degree / norm ----------------
__global__ void k_init_deg(float* __restrict__ deg, int n) {
    int i = blockIdx.x * blockDim.x + threadIdx.x;
    if (i < n) deg[i] = 1.0f;                       // self-loop contribution
}

__global__ void k_deg_edges(const int* __restrict__ dst, float* __restrict__ deg, int e) {
    int i = blockIdx.x * blockDim.x + threadIdx.x;
    if (i < e) atomicAdd(&deg[dst[i]], 1.0f);
}

__global__ void k_rsqrt_inplace(float* __restrict__ deg, int n) {
    int i = blockIdx.x * blockDim.x + threadIdx.x;
    if (i < n) deg[i] = rsqrtf(deg[i]);
}

// ---------------- f32 WMMA GEMM: C[M,NC] = A[M,K] @ B[K,NC] ----------------
// One wave32 computes one 16x16 output tile via V_WMMA_F32_16X16X4_F32.
// A fragment (16x4): lane<16 -> row=lane, K={k0,k0+1}; lane>=16 -> row=lane-16, K={k0+2,k0+3}.
// B fragment (4x16): same striping with N=lane&15.
// C/D: VGPR j holds (M = j + 8*(lane>>4), N = lane&15).
__global__ void k_gemm_f32_wmma(const float* __restrict__ A, const float* __restrict__ B,
                                float* __restrict__ C, int M, int K, int NC) {
    const int lane = threadIdx.x & 31;
    const int wave = blockIdx.x * (blockDim.x >> 5) + (threadIdx.x >> 5);
    const int colTiles = NC >> 4;
    const int tr = wave / colTiles;
    const int tc = wave - tr * colTiles;
    if ((tr << 4) >= M) return;                     // wave-uniform: EXEC stays all-1s

    const int r  = lane & 15;
    const int kk = (lane >> 4) << 1;                // 0 or 2

    const float* arow = A + (size_t)((tr << 4) + r) * K;
    const float* bcol = B + (size_t)((tc << 4) + r);

    v8f acc = {};
    for (int k0 = 0; k0 < K; k0 += 4) {
        v2f av, bv;
        av.x = arow[k0 + kk];
        av.y = arow[k0 + kk + 1];
        bv.x = bcol[(size_t)(k0 + kk) * NC];
        bv.y = bcol[(size_t)(k0 + kk + 1) * NC];
        acc = __builtin_amdgcn_wmma_f32_16x16x4_f32(
            /*neg_a=*/false, av, /*neg_b=*/false, bv,
            /*c_mod=*/(short)0, acc, /*reuse_a=*/false, /*reuse_b=*/false);
    }

    const int mo = (lane >> 4) << 3;                // 0 or 8
    float* cp = C + (size_t)(tc << 4) + (lane & 15);
#pragma unroll
    for (int j = 0; j < 8; ++j)
        cp[(size_t)((tr << 4) + mo + j) * NC] = acc[j];
}

// ---------------- aggregation (HIDDEN == 256 hard-coded: >>8, &255) ----------------
__global__ void k_agg_init_selfloop(const float* __restrict__ h, const float* __restrict__ dinv,
                                    float* __restrict__ agg, int total) {
    int i = blockIdx.x * blockDim.x + threadIdx.x;
    if (i < total) {
        float d = dinv[i >> 8];
        agg[i] = h[i] * d * d;                      // self-loop message, also zero-inits agg
    }
}

__global__ void k_scatter_edges(const float* __restrict__ h, float* __restrict__ agg,
                                const int* __restrict__ src, const int* __restrict__ dst,
                                const float* __restrict__ dinv, long long total) {
    long long tid = (long long)blockIdx.x * blockDim.x + threadIdx.x;
    if (tid >= total) return;
    int e = (int)(tid >> 6);                        // 64 float4 chunks per 256-wide row
    int c = (int)(tid & 63);
    int s = src[e], d = dst[e];
    float nrm = dinv[s] * dinv[d];
    const float4* hv = (const float4*)(h + ((size_t)s << 8));
    float4 v = hv[c];
    float* ap = agg + ((size_t)d << 8) + (c << 2);
    atomicAdd(ap + 0, v.x * nrm);
    atomicAdd(ap + 1, v.y * nrm);
    atomicAdd(ap + 2, v.z * nrm);
    atomicAdd(ap + 3, v.w * nrm);
}

__global__ void k_bias_relu(float* __restrict__ z, const float* __restrict__ b, int total) {
    int i = blockIdx.x * blockDim.x + threadIdx.x;
    if (i < total) z[i] = fmaxf(z[i] + b[i & 255], 0.0f);
}

// ---------------- pooling + head ----------------
__global__ void k_zero(float* __restrict__ p, int n) {
    int i = blockIdx.x * blockDim.x + threadIdx.x;
    if (i < n) p[i] = 0.0f;
}

__global__ void k_pool_sum(const float* __restrict__ z, const int* __restrict__ batch,
                           float* __restrict__ gsum, int total) {
    int i = blockIdx.x * blockDim.x + threadIdx.x;
    if (i < total) {
        int node = i >> 8, j = i & 255;
        atomicAdd(&gsum[(batch[node] << 8) + j], z[i]);
    }
}

__global__ void k_pool_count(const int* __restrict__ batch, float* __restrict__ gcnt, int n) {
    int i = blockIdx.x * blockDim.x + threadIdx.x;
    if (i < n) atomicAdd(&gcnt[batch[i]], 1.0f);
}

__global__ void k_finalize(const float* __restrict__ gsum, const float* __restrict__ gcnt,
                           const float* __restrict__ Wf, const float* __restrict__ bf,
                           float* __restrict__ out) {
    __shared__ float pooled[HIDDEN];
    int g = blockIdx.x;
    float inv = 1.0f / fmaxf(gcnt[g], 1.0f);
    for (int j = threadIdx.x; j < HIDDEN; j += blockDim.x)
        pooled[j] = gsum[(g << 8) + j] * inv;
    __syncthreads();
    if (threadIdx.x < 2 * ACTION_DIM) {
        int j = threadIdx.x;
        float acc = bf[j];
        for (int k = 0; k < HIDDEN; ++k)
            acc += pooled[k] * Wf[k * (2 * ACTION_DIM) + j];
        if (j < ACTION_DIM) {
            out[g * ACTION_DIM + j] = acc;                               // mean
        } else {
            float ls = fminf(fmaxf(acc, -20.0f), 2.0f);
            out[NUM_GRAPHS * ACTION_DIM + g * ACTION_DIM + (j - ACTION_DIM)] = expf(ls);  // std
        }
    }
}

extern "C" void kernel_launch(void* const* d_in, const int* in_sizes, int n_in,
                              void* d_out, int out_size, void* d_ws, size_t ws_size,
                              hipStream_t stream) {
    (void)in_sizes; (void)n_in; (void)out_size; (void)ws_size;
    const float* x   = (const float*)d_in[0];
    const int*   ei  = (const int*)  d_in[1];
    const int*   bat = (const int*)  d_in[2];
    const float* W1  = (const float*)d_in[3];
    const float* b1  = (const float*)d_in[4];
    const float* W2  = (const float*)d_in[5];
    const float* b2  = (const float*)d_in[6];
    const float* Wf  = (const float*)d_in[7];
    const float* bf  = (const float*)d_in[8];
    float* out = (float*)d_out;

    const int N = N_NODES, E = N_EDGES, H = HIDDEN, G = NUM_GRAPHS;
    const int* src = ei;
    const int* dst = ei + E;

    float* ws   = (float*)d_ws;
    float* dinv = ws;                                   // N floats (deg -> rsqrt in place)
    float* h    = ws + 65536;                           // N*H
    float* agg  = h + (size_t)N * H;                    // N*H
    float* gsum = agg + (size_t)N * H;                  // G*H
    float* gcnt = gsum + (size_t)G * H;                 // G (contiguous with gsum)

    const int NH = N * H;                               // 12.8M, fits int
    const long long ET = (long long)E * (H / 4);        // 25.6M scatter threads
    const int tilesPerGemm = (N / 16) * (H / 16);       // 50000 waves
    const int gemmBlocks = tilesPerGemm / 8;            // 8 waves per 256-thread block

    dim3 blk(256);
    // degrees + symmetric norm
    k_init_deg      <<<(N + 255) / 256, blk, 0, stream>>>(dinv, N);
    k_deg_edges     <<<(E + 255) / 256, blk, 0, stream>>>(dst, dinv, E);
    k_rsqrt_inplace <<<(N + 255) / 256, blk, 0, stream>>>(dinv, N);

    // layer 1: h = x @ W1 ; aggregate ; + b1 ; ReLU
    k_gemm_f32_wmma    <<<gemmBlocks, blk, 0, stream>>>(x, W1, h, N, NODE_FEAT, H);
    k_agg_init_selfloop<<<(NH + 255) / 256, blk, 0, stream>>>(h, dinv, agg, NH);
    k_scatter_edges    <<<(int)((ET + 255) / 256), blk, 0, stream>>>(h, agg, src, dst, dinv, ET);
    k_bias_relu        <<<(NH + 255) / 256, blk, 0, stream>>>(agg, b1, NH);

    // layer 2: h = z1 @ W2 ; aggregate ; + b2 ; ReLU   (ping-pong h <-> agg)
    k_gemm_f32_wmma    <<<gemmBlocks, blk, 0, stream>>>(agg, W2, h, N, H, H);
    k_agg_init_selfloop<<<(NH + 255) / 256, blk, 0, stream>>>(h, dinv, agg, NH);
    k_scatter_edges    <<<(int)((ET + 255) / 256), blk, 0, stream>>>(h, agg, src, dst, dinv, ET);
    k_bias_relu        <<<(NH + 255) / 256, blk, 0, stream>>>(agg, b2, NH);

    // global mean pool + policy head
    k_zero      <<<(G * H + G + 255) / 256, blk, 0, stream>>>(gsum, G * H + G);
    k_pool_sum  <<<(NH + 255) / 256, blk, 0, stream>>>(agg, bat, gsum, NH);
    k_pool_count<<<(N + 255) / 256, blk, 0, stream>>>(bat, gcnt, N);
    k_finalize  <<<G, blk, 0, stream>>>(gsum, gcnt, Wf, bf, out);
}